// Expand_8546984919338
// MI455X (gfx1250) — compile-verified
//
#include <hip/hip_runtime.h>

// Expand (ragged repeat-gather): out[b,t,:] = x[b, src(b,t), :] for t < total[b], else 0.
// Pure data movement: ~135 MB HBM traffic -> ~5.8 us roofline at 23.3 TB/s.
//
// Kernel A: per-batch prefix-sum of durations (async global->LDS + LDS scan),
//           then scatter flattened source-row ids into idxbuf[B*T] (-1 = pad).
// Kernel B: coalesced B128 gather from x + non-temporal B128 store of output.

typedef __attribute__((ext_vector_type(4))) float f32x4;

#define SCAN_THREADS 1024

__global__ __launch_bounds__(SCAN_THREADS)
void expand_scan_scatter(const int* __restrict__ dims,
                         int* __restrict__ idxbuf,
                         int S, int T) {
    __shared__ int sm[SCAN_THREADS];
    const int b = blockIdx.x;
    const int i = threadIdx.x;

    if (i < S) {
        // CDNA5 async copy: durations row straight into LDS, no VGPR staging.
        // Tracked by ASYNCcnt; each wave drains its own count before the barrier.
        unsigned lds_off = (unsigned)(size_t)(&sm[i]);   // low 32 bits of flat-LDS addr = LDS byte offset
        const int* gptr  = dims + b * S + i;
        asm volatile("global_load_async_to_lds_b32 %0, %1, off\n\t"
                     "s_wait_asynccnt 0"
                     :: "v"(lds_off), "v"(gptr)
                     : "memory");
    } else {
        sm[i] = 0;
    }
    __syncthreads();

    const int d = sm[i];

    // Hillis-Steele inclusive scan over SCAN_THREADS elements in LDS.
    for (int off = 1; off < SCAN_THREADS; off <<= 1) {
        int v = (i >= off) ? sm[i - off] : 0;
        __syncthreads();
        sm[i] += v;
        __syncthreads();
    }

    const int incl  = sm[i];                  // inclusive cumsum at token i
    const int total = sm[SCAN_THREADS - 1];   // row total (zeros padded above S)
    const int start = incl - d;

    int* __restrict__ row = idxbuf + (size_t)b * (size_t)T;
    const int srcid = b * S + i;              // flattened source row id (bakes in batch)
    for (int t = start; t < incl; ++t)        // <=9 writes per token
        row[t] = srcid;
    for (int t = total + i; t < T; t += SCAN_THREADS)  // pad tail
        row[t] = -1;
}

template <int D4>
__global__ __launch_bounds__(256)
void expand_gather(const f32x4* __restrict__ x4,
                   const int* __restrict__ idxbuf,
                   f32x4* __restrict__ out4,
                   unsigned n4) {
    const unsigned gid = (unsigned)blockIdx.x * 256u + threadIdx.x;
    if (gid >= n4) return;
    const unsigned row = gid / (unsigned)D4;        // D4 compile-time -> mul/shift
    const unsigned e   = gid - row * (unsigned)D4;
    const int s = idxbuf[row];                      // flattened source row, -1 = pad
    f32x4 v = {0.f, 0.f, 0.f, 0.f};
    if (s >= 0)
        v = x4[(unsigned)s * (unsigned)D4 + e];     // global_load_b128, stays temporal (reused ~4.5x)
    __builtin_nontemporal_store(v, out4 + gid);     // streaming write-once: NT b128 store
}

__global__ __launch_bounds__(256)
void expand_gather_dyn(const f32x4* __restrict__ x4,
                       const int* __restrict__ idxbuf,
                       f32x4* __restrict__ out4,
                       unsigned n4, int D4) {
    const unsigned gid = (unsigned)blockIdx.x * 256u + threadIdx.x;
    if (gid >= n4) return;
    const unsigned row = gid / (unsigned)D4;
    const unsigned e   = gid - row * (unsigned)D4;
    const int s = idxbuf[row];
    f32x4 v = {0.f, 0.f, 0.f, 0.f};
    if (s >= 0)
        v = x4[(unsigned)s * (unsigned)D4 + e];
    __builtin_nontemporal_store(v, out4 + gid);
}

extern "C" void kernel_launch(void* const* d_in, const int* in_sizes, int n_in,
                              void* d_out, int out_size, void* d_ws, size_t ws_size,
                              hipStream_t stream) {
    const float* x  = (const float*)d_in[0];     // [B, S, D] f32
    const int* dims = (const int*)d_in[1];       // [B, S] i32

    const int B  = 16;                           // per reference setup
    const int S  = in_sizes[1] / B;              // 1024
    const int D  = in_sizes[0] / in_sizes[1];    // 384
    const int D4 = D / 4;                        // 96
    const int T  = out_size / (B * D);           // output length (harness-sized)

    int* idxbuf = (int*)d_ws;                    // B*T ints (~300 KB)

    expand_scan_scatter<<<B, SCAN_THREADS, 0, stream>>>(dims, idxbuf, S, T);

    const unsigned n4     = (unsigned)(out_size / 4);
    const unsigned blocks = (n4 + 255u) / 256u;
    if (D4 == 96) {
        expand_gather<96><<<blocks, 256, 0, stream>>>(
            (const f32x4*)x, idxbuf, (f32x4*)d_out, n4);
    } else {
        expand_gather_dyn<<<blocks, 256, 0, stream>>>(
            (const f32x4*)x, idxbuf, (f32x4*)d_out, n4, D4);
    }
}